// SynthesisConstBlock_9234179686689
// MI455X (gfx1250) — compile-verified
//
#include <hip/hip_runtime.h>

typedef __attribute__((ext_vector_type(2))) float v2f;
typedef __attribute__((ext_vector_type(4))) float v4f;
typedef __attribute__((ext_vector_type(8))) float v8f;

#define BATCH 64
#define CCH   512
#define WDIM  512
#define NPOS  16            // RES*RES
#define NCOL  (CCH * NPOS)  // 8192
#define MT    4             // all of M=64 as 4 x 16-row tiles per wave

__device__ __forceinline__ v8f wmma_f32_16x16x4(v2f a, v2f b, v8f c) {
  // 8 args: (neg_a, A, neg_b, B, c_mod, C, reuse_a, reuse_b)
  return __builtin_amdgcn_wmma_f32_16x16x4_f32(false, a, false, b, (short)0, c,
                                               false, false);
}

// ---------------------------------------------------------------------------
// Kernel 1: style GEMM  s[b,c] = (w0 @ mod_w) / sqrt(512) + mod_b + 1
// grid (CCH/16), block 32: one wave owns a 64x16 slab (all 4 M-tiles), so
// each mod_w element is fetched exactly once and feeds 4 WMMAs.
// ---------------------------------------------------------------------------
__global__ void style_gemm_kernel(const float* __restrict__ w0,
                                  const float* __restrict__ mod_w,
                                  const float* __restrict__ mod_b,
                                  float* __restrict__ s_out) {
  const int lane  = threadIdx.x;
  const int n0    = blockIdx.x * 16;
  const int col   = lane & 15;           // row-in-tile for A, col for B/C
  const int khalf = (lane >> 4) * 2;

  v8f acc[MT] = {};
  for (int k0 = 0; k0 < WDIM; k0 += 4) {
    v2f b;
    b.x = mod_w[(size_t)(k0 + khalf) * CCH + n0 + col];
    b.y = mod_w[(size_t)(k0 + khalf + 1) * CCH + n0 + col];
    v2f a[MT];
#pragma unroll
    for (int mt = 0; mt < MT; ++mt)
      a[mt] = *(const v2f*)(w0 + (size_t)(mt * 16 + col) * WDIM + k0 + khalf);
#pragma unroll
    for (int mt = 0; mt < MT; ++mt)
      acc[mt] = wmma_f32_16x16x4(a[mt], b, acc[mt]);
  }
  const float inv = 0.04419417382415922f;  // 1/sqrt(512)
  const float bb  = mod_b[n0 + col] + 1.0f;
#pragma unroll
  for (int mt = 0; mt < MT; ++mt) {
#pragma unroll
    for (int r = 0; r < 8; ++r) {
      int m = mt * 16 + r + 8 * (lane >> 4);
      s_out[(size_t)m * CCH + n0 + col] = acc[mt][r] * inv + bb;
    }
  }
}

// ---------------------------------------------------------------------------
// Kernel 2: batch-independent precompute
//   T[i, o*16+p] = sum_taps w_rt[ky,kx,i,o] * const_pad[i, p(+tap)]
//   Q[i, o]      = sum_taps w_rt[ky,kx,i,o]^2
// one thread per (i,o) pair; conv_w is HWIO, const is [1,C,4,4]
// ---------------------------------------------------------------------------
__global__ void precomp_kernel(const float* __restrict__ conv_w,
                               const float* __restrict__ cst,
                               float* __restrict__ T,
                               float* __restrict__ Q) {
  const int t = blockIdx.x * blockDim.x + threadIdx.x;
  const int o = t & (CCH - 1);
  const int i = t >> 9;
  const float scale = 0.014731391274719736f;  // 1/sqrt(3*3*512)

  float cw[9];
  float q = 0.f;
#pragma unroll
  for (int tap = 0; tap < 9; ++tap) {
    float w = conv_w[((size_t)tap * CCH + i) * CCH + o];
    cw[tap] = w;
    q += w * w;
  }
  Q[(size_t)i * CCH + o] = q * scale * scale;

  float cv[16];
#pragma unroll
  for (int p = 0; p < 16; ++p) cv[p] = cst[i * 16 + p];

  float outv[16];
#pragma unroll
  for (int py = 0; py < 4; ++py) {
#pragma unroll
    for (int px = 0; px < 4; ++px) {
      float acc = 0.f;
#pragma unroll
      for (int ky = 0; ky < 3; ++ky) {
        int yy = py + ky - 1;
        if (yy < 0 || yy > 3) continue;
#pragma unroll
        for (int kx = 0; kx < 3; ++kx) {
          int xx = px + kx - 1;
          if (xx < 0 || xx > 3) continue;
          acc += cw[ky * 3 + kx] * cv[yy * 4 + xx];
        }
      }
      outv[py * 4 + px] = acc * scale;
    }
  }
  float* dst = T + (size_t)i * NCOL + o * 16;
#pragma unroll
  for (int v = 0; v < 4; ++v)
    *(v4f*)(dst + 4 * v) = *(const v4f*)(outv + 4 * v);
}

// ---------------------------------------------------------------------------
// Kernel 3: demod GEMM  d[b,o] = rsqrt( (s*s) @ Q + 1e-8 )
// grid (CCH/16), block 32: full-M slab per wave, Q fetched once.
// ---------------------------------------------------------------------------
__global__ void demod_gemm_kernel(const float* __restrict__ s,
                                  const float* __restrict__ Q,
                                  float* __restrict__ d_out) {
  const int lane  = threadIdx.x;
  const int n0    = blockIdx.x * 16;
  const int col   = lane & 15;
  const int khalf = (lane >> 4) * 2;

  v8f acc[MT] = {};
  for (int k0 = 0; k0 < CCH; k0 += 4) {
    v2f b;
    b.x = Q[(size_t)(k0 + khalf) * CCH + n0 + col];
    b.y = Q[(size_t)(k0 + khalf + 1) * CCH + n0 + col];
    v2f a[MT];
#pragma unroll
    for (int mt = 0; mt < MT; ++mt) {
      a[mt] = *(const v2f*)(s + (size_t)(mt * 16 + col) * CCH + k0 + khalf);
      a[mt].x *= a[mt].x;
      a[mt].y *= a[mt].y;
    }
#pragma unroll
    for (int mt = 0; mt < MT; ++mt)
      acc[mt] = wmma_f32_16x16x4(a[mt], b, acc[mt]);
  }
#pragma unroll
  for (int mt = 0; mt < MT; ++mt) {
#pragma unroll
    for (int r = 0; r < 8; ++r) {
      int m = mt * 16 + r + 8 * (lane >> 4);
      d_out[(size_t)m * CCH + n0 + col] = __frsqrt_rn(acc[mt][r] + 1e-8f);
    }
  }
}

// ---------------------------------------------------------------------------
// Kernel 4: main GEMM + fused epilogue
//   y[b, o*16+p] = lrelu( d[b,o] * (s @ T)[b, o*16+p] + noise*ns + bias[o] ) * sqrt(2)
// grid (CCH), block 32: one wave owns a full 64x16 output slab for one output
// channel o (a 16-wide N tile == 16 spatial positions of channel o). T is
// streamed exactly once; each pair of B loads feeds 4 WMMAs.
// ---------------------------------------------------------------------------
__global__ void modconv_gemm_kernel(const float* __restrict__ s,
                                    const float* __restrict__ T,
                                    const float* __restrict__ dmod,
                                    const float* __restrict__ noise,
                                    const float* __restrict__ ns_ptr,
                                    const float* __restrict__ bias,
                                    float* __restrict__ out) {
  const int lane  = threadIdx.x;
  const int o     = blockIdx.x;
  const int n0    = o * 16;
  const int col   = lane & 15;           // spatial position p
  const int khalf = (lane >> 4) * 2;

  v8f acc[MT] = {};
  for (int k0 = 0; k0 < CCH; k0 += 4) {
    const float* brow = T + (size_t)(k0 + khalf) * NCOL + n0 + col;
    v2f b;
    b.x = brow[0];
    b.y = brow[NCOL];
    if (k0 + 32 < CCH)  // gfx1250 global_prefetch_b8 on the streamed T rows
      __builtin_prefetch(brow + (size_t)32 * NCOL, 0, 1);
    v2f a[MT];
#pragma unroll
    for (int mt = 0; mt < MT; ++mt)
      a[mt] = *(const v2f*)(s + (size_t)(mt * 16 + col) * CCH + k0 + khalf);
#pragma unroll
    for (int mt = 0; mt < MT; ++mt)
      acc[mt] = wmma_f32_16x16x4(a[mt], b, acc[mt]);
  }

  const float ns = *ns_ptr;
  const float bo = bias[o];
#pragma unroll
  for (int mt = 0; mt < MT; ++mt) {
#pragma unroll
    for (int r = 0; r < 8; ++r) {
      int b_idx = mt * 16 + r + 8 * (lane >> 4);
      float v = acc[mt][r] * dmod[(size_t)b_idx * CCH + o] +
                noise[b_idx * NPOS + col] * ns + bo;
      v = (v > 0.f ? v : 0.2f * v) * 1.4142135623730951f;
      out[((size_t)b_idx * CCH + o) * NPOS + col] = v;
    }
  }
}

// ---------------------------------------------------------------------------
extern "C" void kernel_launch(void* const* d_in, const int* in_sizes, int n_in,
                              void* d_out, int out_size, void* d_ws,
                              size_t ws_size, hipStream_t stream) {
  (void)in_sizes; (void)n_in; (void)out_size; (void)ws_size;

  const float* w0    = (const float*)d_in[0];  // [64,512]
  const float* cst   = (const float*)d_in[1];  // [1,512,4,4]
  const float* convw = (const float*)d_in[2];  // [3,3,512,512] HWIO
  const float* modw  = (const float*)d_in[3];  // [512,512]
  const float* modb  = (const float*)d_in[4];  // [512]
  const float* noise = (const float*)d_in[5];  // [64,1,4,4]
  const float* nstr  = (const float*)d_in[6];  // scalar
  const float* bias  = (const float*)d_in[7];  // [512]
  float* out = (float*)d_out;

  char* ws = (char*)d_ws;
  float* T = (float*)(ws);                         // 512*8192*4 = 16 MB
  float* Q = (float*)(ws + (size_t)16777216);      // 512*512*4  =  1 MB
  float* S = (float*)(ws + (size_t)17825792);      // 64*512*4   = 128 KB
  float* D = (float*)(ws + (size_t)17956864);      // 64*512*4   = 128 KB

  // style GEMM: s = w0 @ mod_w' + mod_b + 1
  style_gemm_kernel<<<dim3(CCH / 16), 32, 0, stream>>>(w0, modw, modb, S);
  // batch-independent conv precompute: T, Q
  precomp_kernel<<<dim3((CCH * CCH) / 256), 256, 0, stream>>>(convw, cst, T, Q);
  // demod GEMM: d = rsqrt(s^2 @ Q + eps)
  demod_gemm_kernel<<<dim3(CCH / 16), 32, 0, stream>>>(S, Q, D);
  // main GEMM + fused noise/bias/lrelu epilogue
  modconv_gemm_kernel<<<dim3(CCH), 32, 0, stream>>>(S, T, D, noise, nstr, bias,
                                                    out);
}